// AutoEncoder_35278861369470
// MI455X (gfx1250) — compile-verified
//
#include <hip/hip_runtime.h>
#include <math.h>

#define BB 32
#define LL 48
#define XS 256
#define HS 129           // XSIZE/2 + 1
#define KDIM 52          // 48 latent + 3 coord + 1 const

typedef float v2f __attribute__((ext_vector_type(2)));
typedef float v8f __attribute__((ext_vector_type(8)));

// ---------------------------------------------------------------- zero decoded
__global__ void k_zero(float* __restrict__ p, int n) {
  int i = blockIdx.x * blockDim.x + threadIdx.x;
  if (i < n) p[i] = 0.0f;
}

// ---------------------------------------------------------------- per-row setup
// Aext[row=2b+i][0..47]  = R2[b,i,:] @ latents[b,:,:]   (rotated latents)
// Aext[row][48..50]      = R2[b,i,:]                    (applies to coords)
// Aext[row][51]          = shift + XSIZE/2              (const column)
__global__ void k_setup(const float* __restrict__ lx, const float* __restrict__ ly,
                        const float* __restrict__ lz, const float* __restrict__ dEuler,
                        const float* __restrict__ dShift, const int* __restrict__ idxs,
                        const float* __restrict__ rotT, const float* __restrict__ tiltT,
                        const float* __restrict__ psiT, const float* __restrict__ sxT,
                        const float* __restrict__ syT, float* __restrict__ Aext) {
  int row = threadIdx.x;
  if (row >= 2 * BB) return;
  int b = row >> 1, i = row & 1;
  int idx = idxs[b];
  float rot  = rotT[idx]  + dEuler[b * 3 + 0];
  float tilt = tiltT[idx] + dEuler[b * 3 + 1];
  float psi  = psiT[idx]  + dEuler[b * 3 + 2];
  float ca = cosf(rot),  sa = sinf(rot);
  float cb = cosf(tilt), sb = sinf(tilt);
  float cg = cosf(psi),  sg = sinf(psi);
  float r0, r1, r2;
  if (i == 0) { r0 =  cg*cb*ca - sg*sa; r1 =  cg*cb*sa + sg*ca; r2 = -cg*sb; }
  else        { r0 = -sg*cb*ca - cg*sa; r1 = -sg*cb*sa + cg*ca; r2 =  sg*sb; }
  float shift = (i == 0) ? (sxT[idx] + dShift[b * 2 + 0])
                         : (syT[idx] + dShift[b * 2 + 1]);
  float* Ar = Aext + row * KDIM;
  for (int l = 0; l < LL; ++l)
    Ar[l] = r0 * lx[b * LL + l] + r1 * ly[b * LL + l] + r2 * lz[b * LL + l];
  Ar[48] = r0; Ar[49] = r1; Ar[50] = r2;
  Ar[51] = shift + 0.5f * XS;
}

// ---------------------------------------------------------------- CTF
__global__ void k_ctf(const int* __restrict__ idxs, const float* __restrict__ dU,
                      const float* __restrict__ dV, const float* __restrict__ dA,
                      const float* __restrict__ csT, float* __restrict__ ctf_out) {
  int t = blockIdx.x * blockDim.x + threadIdx.x;
  const int per_b = XS * HS;
  if (t >= BB * per_b) return;
  int b  = t / per_b;
  int r  = t - b * per_b;
  int ky = r / HS;
  int kx = r - ky * HS;
  int idx = idxs[b];
  const float u  = 300.0f * 1000.0f;                       // KV * 1000
  const float wl = 12.2643247f / sqrtf(u * (1.0f + u * 9.78466e-07f));
  float fy = (ky < XS / 2 ? (float)ky : (float)(ky - XS)) * (1.0f / XS);
  float fx = (float)kx * (1.0f / XS);
  float f2 = fy * fy + fx * fx;
  float ang = atan2f(fy, fx);
  float du = dU[idx], dv = dV[idx], da = dA[idx];
  float csA = csT[idx] * 1.0e7f;
  float defocus = 0.5f * (du + dv + (du - dv) * cosf(2.0f * (ang - da)));
  const float PIf = 3.14159265358979f;
  float chi = PIf * wl * defocus * f2 - 0.5f * PIf * csA * wl * wl * wl * f2 * f2;
  ctf_out[t] = -sinf(chi);
}

// ---------------------------------------------------------------- WMMA proj + scatter
__device__ __forceinline__ void scat_tile(v8f a, int t, int bbase, float v,
                                          float* __restrict__ dec) {
#pragma unroll
  for (int k = 0; k < 4; ++k) {
    int b = t * 8 + bbase + k;
    float px = a[2 * k], py = a[2 * k + 1];
    int ix = __float2int_rn(px); ix = ix < 0 ? 0 : (ix > XS - 1 ? XS - 1 : ix);
    int iy = __float2int_rn(py); iy = iy < 0 ? 0 : (iy > XS - 1 ? XS - 1 : iy);
    atomicAdd(dec + ((size_t)b << 16) + (iy << 8) + ix, v);
  }
}

// 8 waves per block; wave w owns 16-column strip n0 = blockIdx*128 + w*16.
// Columns with n >= N compute garbage from clamped addresses (branch-free)
// and are simply skipped at scatter time — WMMA needs EXEC all-ones anyway.
__global__ __launch_bounds__(256)
void k_proj_scatter(const float* __restrict__ Aext, const float* __restrict__ Z,
                    const float* __restrict__ coords, const float* __restrict__ values,
                    float* __restrict__ decoded, int N) {
  __shared__ float As[64 * KDIM];
  for (int i = threadIdx.x; i < 64 * KDIM; i += 256) As[i] = Aext[i];
  __syncthreads();

  int wave = threadIdx.x >> 5;
  int lane = threadIdx.x & 31;
  bool hi  = lane >= 16;
  int nl   = lane & 15;
  int n    = blockIdx.x * 128 + wave * 16 + nl;
  int n_cl = n < N ? n : (N - 1);             // clamp, don't branch

  // preload coords + value early; latency hidden behind the WMMA loop
  float c0 = coords[(size_t)n_cl * 3 + 0];
  float c1 = coords[(size_t)n_cl * 3 + 1];
  float c2 = coords[(size_t)n_cl * 3 + 2];
  float val = values[n_cl];

  const float2* Zv = (const float2*)Z;        // Z row = 24 float2, 8B-aligned
  size_t zrow = (size_t)n_cl * (LL / 2);

  v8f acc0 = 0.0f, acc1 = 0.0f, acc2 = 0.0f, acc3 = 0.0f;

#pragma unroll
  for (int k0 = 0; k0 < LL; k0 += 4) {
    int kb = k0 + (hi ? 2 : 0);
    float2 bz = Zv[zrow + (kb >> 1)];          // one global_load_b64 per k-step
    v2f bv; bv.x = bz.x; bv.y = bz.y;
    v2f a0, a1, a2, a3;
    a0.x = As[(0 * 16 + nl) * KDIM + kb]; a0.y = As[(0 * 16 + nl) * KDIM + kb + 1];
    a1.x = As[(1 * 16 + nl) * KDIM + kb]; a1.y = As[(1 * 16 + nl) * KDIM + kb + 1];
    a2.x = As[(2 * 16 + nl) * KDIM + kb]; a2.y = As[(2 * 16 + nl) * KDIM + kb + 1];
    a3.x = As[(3 * 16 + nl) * KDIM + kb]; a3.y = As[(3 * 16 + nl) * KDIM + kb + 1];
    acc0 = __builtin_amdgcn_wmma_f32_16x16x4_f32(false, a0, false, bv, (short)0, acc0, false, false);
    acc1 = __builtin_amdgcn_wmma_f32_16x16x4_f32(false, a1, false, bv, (short)0, acc1, false, false);
    acc2 = __builtin_amdgcn_wmma_f32_16x16x4_f32(false, a2, false, bv, (short)0, acc2, false, false);
    acc3 = __builtin_amdgcn_wmma_f32_16x16x4_f32(false, a3, false, bv, (short)0, acc3, false, false);
  }

  // final k-step: K = 48..51  ->  [coords.x, coords.y | coords.z, 1.0]
  {
    const int kb = LL + (hi ? 2 : 0);
    v2f bv;
    bv.x = hi ? c2 : c0;                       // uniform per half-wave: v_cndmask
    bv.y = hi ? 1.0f : c1;
    v2f a0, a1, a2, a3;
    a0.x = As[(0 * 16 + nl) * KDIM + kb]; a0.y = As[(0 * 16 + nl) * KDIM + kb + 1];
    a1.x = As[(1 * 16 + nl) * KDIM + kb]; a1.y = As[(1 * 16 + nl) * KDIM + kb + 1];
    a2.x = As[(2 * 16 + nl) * KDIM + kb]; a2.y = As[(2 * 16 + nl) * KDIM + kb + 1];
    a3.x = As[(3 * 16 + nl) * KDIM + kb]; a3.y = As[(3 * 16 + nl) * KDIM + kb + 1];
    acc0 = __builtin_amdgcn_wmma_f32_16x16x4_f32(false, a0, false, bv, (short)0, acc0, false, false);
    acc1 = __builtin_amdgcn_wmma_f32_16x16x4_f32(false, a1, false, bv, (short)0, acc1, false, false);
    acc2 = __builtin_amdgcn_wmma_f32_16x16x4_f32(false, a2, false, bv, (short)0, acc2, false, false);
    acc3 = __builtin_amdgcn_wmma_f32_16x16x4_f32(false, a3, false, bv, (short)0, acc3, false, false);
  }

  if (n < N) {
    int bbase = hi ? 4 : 0;
    scat_tile(acc0, 0, bbase, val, decoded);
    scat_tile(acc1, 1, bbase, val, decoded);
    scat_tile(acc2, 2, bbase, val, decoded);
    scat_tile(acc3, 3, bbase, val, decoded);
  }
}

// ---------------------------------------------------------------- 256-pt LDS FFT
// Twiddles: all butterflies use exp(-i*pi*j/128), j = pos << (7-stage).
// One 128-entry LDS table per block replaces 8 sincosf per thread with 8 LDS loads.
__device__ __forceinline__ void tw_build(float2* tw, int tid) {
  // tid in [0,128): one accurate sincosf per thread, once per block
  float ang = -3.14159265358979f * (float)tid * (1.0f / 128.0f);
  float sn, cs;
  sincosf(ang, &sn, &cs);
  tw[tid] = make_float2(cs, sn);
}

__device__ __forceinline__ void fft256(float2* s, const float2* tw, int tid,
                                       bool inverse) {
  for (int i = tid; i < 256; i += 128) {
    int j = (int)(__brev((unsigned)i) >> 24);
    if (i < j) { float2 t = s[i]; s[i] = s[j]; s[j] = t; }
  }
  __syncthreads();
#pragma unroll
  for (int st = 0; st < 8; ++st) {
    int h   = 1 << st;
    int pos = tid & (h - 1);
    int i0  = ((tid >> st) << (st + 1)) + pos;
    int i1  = i0 + h;
    float2 w = tw[pos << (7 - st)];            // ds_load_b64
    float cs = w.x;
    float sn = inverse ? -w.y : w.y;           // conj for inverse
    float2 a = s[i0], b = s[i1];
    float tr = b.x * cs - b.y * sn;
    float ti = b.x * sn + b.y * cs;
    s[i0] = make_float2(a.x + tr, a.y + ti);
    s[i1] = make_float2(a.x - tr, a.y - ti);
    __syncthreads();
  }
}

__global__ __launch_bounds__(128)
void k_fft_rows_fwd(const float* __restrict__ dec, float2* __restrict__ spec) {
  __shared__ float2 s[256];
  __shared__ float2 tw[128];
  int blk = blockIdx.x;                 // b*256 + y
  int tid = threadIdx.x;
  tw_build(tw, tid);
  const float* row = dec + (size_t)blk * XS;
  for (int i = tid; i < 256; i += 128) s[i] = make_float2(row[i], 0.0f);
  __syncthreads();
  fft256(s, tw, tid, false);
  float2* out = spec + (size_t)blk * HS;
  for (int i = tid; i < HS; i += 128) out[i] = s[i];
}

__global__ __launch_bounds__(128)
void k_fft_cols(float2* __restrict__ spec, const float* __restrict__ ctf) {
  __shared__ float2 s[256];
  __shared__ float2 tw[128];
  int blk = blockIdx.x;                 // b*HS + kx
  int b   = blk / HS;
  int kx  = blk - b * HS;
  int tid = threadIdx.x;
  tw_build(tw, tid);
  float2* col = spec + (size_t)b * XS * HS + kx;
  for (int i = tid; i < 256; i += 128) s[i] = col[(size_t)i * HS];
  __syncthreads();
  fft256(s, tw, tid, false);            // forward over y
  const float* cp = ctf + (size_t)b * XS * HS + kx;
  for (int i = tid; i < 256; i += 128) {
    float c = cp[(size_t)i * HS];
    s[i].x *= c; s[i].y *= c;
  }
  __syncthreads();
  fft256(s, tw, tid, true);             // inverse over y
  const float inv = 1.0f / 256.0f;
  for (int i = tid; i < 256; i += 128)
    col[(size_t)i * HS] = make_float2(s[i].x * inv, s[i].y * inv);
}

__global__ __launch_bounds__(128)
void k_fft_rows_inv(const float2* __restrict__ spec, float* __restrict__ out) {
  __shared__ float2 s[256];
  __shared__ float2 tw[128];
  int blk = blockIdx.x;                 // b*256 + y
  int tid = threadIdx.x;
  tw_build(tw, tid);
  const float2* row = spec + (size_t)blk * HS;
  for (int i = tid; i < 256; i += 128) {
    if (i < HS) s[i] = row[i];
    else { float2 c = row[256 - i]; s[i] = make_float2(c.x, -c.y); }  // Hermitian
  }
  __syncthreads();
  fft256(s, tw, tid, true);             // inverse over x
  float* orow = out + (size_t)blk * XS;
  const float inv = 1.0f / 256.0f;
  for (int i = tid; i < 256; i += 128) orow[i] = s[i].x * inv;
}

// ---------------------------------------------------------------- launcher
extern "C" void kernel_launch(void* const* d_in, const int* in_sizes, int n_in,
                              void* d_out, int out_size, void* d_ws, size_t ws_size,
                              hipStream_t stream) {
  const float* latent_x     = (const float*)d_in[0];
  const float* latent_y     = (const float*)d_in[1];
  const float* latent_z     = (const float*)d_in[2];
  const float* delta_euler  = (const float*)d_in[3];
  const float* delta_shifts = (const float*)d_in[4];
  const int*   indexes      = (const int*)  d_in[5];
  const float* Z            = (const float*)d_in[6];
  const float* coords       = (const float*)d_in[7];
  const float* values       = (const float*)d_in[8];
  const float* angle_rot    = (const float*)d_in[9];
  const float* angle_tilt   = (const float*)d_in[10];
  const float* angle_psi    = (const float*)d_in[11];
  const float* shift_x_tab  = (const float*)d_in[12];
  const float* shift_y_tab  = (const float*)d_in[13];
  const float* defocusU     = (const float*)d_in[14];
  const float* defocusV     = (const float*)d_in[15];
  const float* defocusAngle = (const float*)d_in[16];
  const float* cs           = (const float*)d_in[17];
  int N = in_sizes[8];                       // values -> N

  float* out         = (float*)d_out;
  float* decoded     = out;                                // [B,256,256]
  float* decoded_ctf = out + (size_t)BB * XS * XS;         // [B,256,256]
  float* ctf         = out + (size_t)2 * BB * XS * XS;     // [B,256,129]

  // workspace: Aext (64x52 f32) then spectrum (B*256*129 float2, ~8.45 MB)
  float*  Aext = (float*)d_ws;
  float2* spec = (float2*)((char*)d_ws + ((64 * KDIM * sizeof(float) + 255) & ~255u));

  int decN = BB * XS * XS;
  k_zero<<<(decN + 255) / 256, 256, 0, stream>>>(decoded, decN);
  k_setup<<<1, 64, 0, stream>>>(latent_x, latent_y, latent_z, delta_euler, delta_shifts,
                                indexes, angle_rot, angle_tilt, angle_psi,
                                shift_x_tab, shift_y_tab, Aext);
  int ctfN = BB * XS * HS;
  k_ctf<<<(ctfN + 255) / 256, 256, 0, stream>>>(indexes, defocusU, defocusV,
                                                defocusAngle, cs, ctf);
  k_proj_scatter<<<(N + 127) / 128, 256, 0, stream>>>(Aext, Z, coords, values, decoded, N);
  k_fft_rows_fwd<<<BB * XS, 128, 0, stream>>>(decoded, spec);
  k_fft_cols<<<BB * HS, 128, 0, stream>>>(spec, ctf);
  k_fft_rows_inv<<<BB * XS, 128, 0, stream>>>(spec, decoded_ctf);
}